// Control_Contrastive_37967510896830
// MI455X (gfx1250) — compile-verified
//
#include <hip/hip_runtime.h>

#define B_N 8192
#define C_N 5994
#define D_N 192

#define COS_M 0.9800665778412416f
#define SIN_M 0.19866933079506122f
#define TH_C  0.9800665778412416f
#define MM_C  0.039733866159012245f
#define S_SC  30.0f
#define EPS_LS 0.1f

typedef __attribute__((ext_vector_type(2))) float v2f;
typedef __attribute__((ext_vector_type(8))) float v8f;

__device__ __forceinline__ void atomicMaxF(float* addr, float val) {
    int old = __float_as_int(*addr);
    while (__int_as_float(old) < val) {
        int prev = atomicCAS((int*)addr, old, __float_as_int(val));
        if (prev == old) break;
        old = prev;
    }
}
__device__ __forceinline__ void atomicMinF(float* addr, float val) {
    int old = __float_as_int(*addr);
    while (__int_as_float(old) > val) {
        int prev = atomicCAS((int*)addr, old, __float_as_int(val));
        if (prev == old) break;
        old = prev;
    }
}

// ---------------------------------------------------------------- init
__global__ void init_ws(float* __restrict__ mx, float* __restrict__ cnt,
                        float* __restrict__ sc) {
    int idx = blockIdx.x * 256 + threadIdx.x;
    if (idx < C_N * D_N) mx[idx] = 0.f;
    if (idx < C_N) cnt[idx] = 0.f;
    if (idx == 0) {
        sc[0] = -__builtin_inff();  // vmax (offdiag G)
        sc[1] =  __builtin_inff();  // gmin (offdiag G)
        sc[2] = 0.f;                // sum1
        sc[3] = 0.f;                // sum2
        sc[4] = 0.f;                // ce_sample sum
        sc[5] = 0.f;                // ce_center sum
    }
}

// ------------------------------------------------------- row normalize
__global__ __launch_bounds__(256) void normalize_rows(const float* __restrict__ in,
                                                      float* __restrict__ out) {
    __shared__ float red[256];
    const int row = blockIdx.x;
    const int t = threadIdx.x;
    float v = 0.f;
    if (t < D_N) v = in[(size_t)row * D_N + t];
    red[t] = v * v;
    __syncthreads();
    for (int s = 128; s > 0; s >>= 1) {
        if (t < s) red[t] += red[t + s];
        __syncthreads();
    }
    float inv = 1.f / fmaxf(sqrtf(red[0]), 1e-12f);
    if (t < D_N) out[(size_t)row * D_N + t] = v * inv;
}

// ------------------------------------------------------- segment mean
__global__ void scatter_mean(const float* __restrict__ x1, const int* __restrict__ label,
                             float* __restrict__ mx, float* __restrict__ cnt) {
    int idx = blockIdx.x * 256 + threadIdx.x;
    if (idx >= B_N * D_N) return;
    int i = idx / D_N;
    int d = idx - i * D_N;
    int l = label[i];
    atomicAdd(&mx[(size_t)l * D_N + d], x1[idx]);
    if (d == 0) atomicAdd(&cnt[l], 1.f);
}

__global__ void div_mean(float* __restrict__ mx, const float* __restrict__ cnt) {
    int idx = blockIdx.x * 256 + threadIdx.x;
    if (idx < C_N * D_N) mx[idx] /= cnt[idx / D_N];
}

// ----------------------------------------------------------- WMMA GEMM
// MODE 0: out = arcface(A·Bᵀ, one_hot = (label[m]==n)) * S   (sample)
// MODE 1: out = arcface(A·Bᵀ, one_hot = (m==n)) * S          (center)
// MODE 2: out = 2*(A·Bᵀ); fused offdiag max/min -> sc[0]/sc[1] (G)
// Block = 8 waves as 2(M)x4(N); wave tile 16x64; K steps of 4.
// Out-of-range tile rows/cols are clamped to row 0 for the loads (always
// in-bounds, finite data); WMMA outputs are independent dot products, so
// garbage lands only in elements the bounds-checked epilogue never stores.
// => K-loop is branchless: pure global_load_b64 + v_wmma.
template <int MODE>
__global__ __launch_bounds__(256) void wmma_gemm(
    const float* __restrict__ A, const float* __restrict__ Bm,
    float* __restrict__ Out, const int* __restrict__ label,
    float* __restrict__ sc, int Mrows, int Ncols) {
    const int lane = threadIdx.x & 31;
    const int wv = threadIdx.x >> 5;
    const int half = lane >> 4;
    const int l16 = lane & 15;
    const int Mb = blockIdx.y * 32 + (wv >> 2) * 16;
    const int Nb = blockIdx.x * 256 + (wv & 3) * 64;

    // A fragment: lanes 0-15 hold K{k0,k0+1}; lanes 16-31 K{k0+2,k0+3}
    const int am = Mb + l16;
    const float* ap = A + (size_t)(am < Mrows ? am : 0) * D_N + 2 * half;

    // B fragments: four separate named pointers so address-space inference
    // keeps them GLOBAL (an array of pointers degraded these to FLAT loads,
    // which cost DScnt + LDS-path arbitration on top of LOADcnt).
    const int n0 = Nb + l16;
    const int n1 = n0 + 16;
    const int n2 = n0 + 32;
    const int n3 = n0 + 48;
    const float* b0p = Bm + (size_t)(n0 < Ncols ? n0 : 0) * D_N + 2 * half;
    const float* b1p = Bm + (size_t)(n1 < Ncols ? n1 : 0) * D_N + 2 * half;
    const float* b2p = Bm + (size_t)(n2 < Ncols ? n2 : 0) * D_N + 2 * half;
    const float* b3p = Bm + (size_t)(n3 < Ncols ? n3 : 0) * D_N + 2 * half;

    v8f acc0 = {}, acc1 = {}, acc2 = {}, acc3 = {};
#pragma unroll 4
    for (int k0 = 0; k0 < D_N; k0 += 4) {
        v2f a = *(const v2f*)ap;
        v2f b0 = *(const v2f*)b0p;
        v2f b1 = *(const v2f*)b1p;
        v2f b2 = *(const v2f*)b2p;
        v2f b3 = *(const v2f*)b3p;
        acc0 = __builtin_amdgcn_wmma_f32_16x16x4_f32(false, a, false, b0,
                                                     (short)0, acc0, false, false);
        acc1 = __builtin_amdgcn_wmma_f32_16x16x4_f32(false, a, false, b1,
                                                     (short)0, acc1, false, false);
        acc2 = __builtin_amdgcn_wmma_f32_16x16x4_f32(false, a, false, b2,
                                                     (short)0, acc2, false, false);
        acc3 = __builtin_amdgcn_wmma_f32_16x16x4_f32(false, a, false, b3,
                                                     (short)0, acc3, false, false);
        ap += 4;
        b0p += 4; b1p += 4; b2p += 4; b3p += 4;
    }

    // Epilogue. C/D layout: VGPR r -> M = Mb + 8*half + r, N = Nb + j*16 + l16
    float lmax = -__builtin_inff(), lmin = __builtin_inff();
#pragma unroll
    for (int r = 0; r < 8; r++) {
        const int m = Mb + half * 8 + r;
        const bool mok = (m < Mrows);
        int lab = 0;
        if (MODE == 0 && mok) lab = label[m];
#pragma unroll
        for (int j = 0; j < 4; j++) {
            const int n = Nb + j * 16 + l16;
            if (mok && n < Ncols) {
                float c = (j == 0) ? acc0[r] : (j == 1) ? acc1[r]
                                             : (j == 2) ? acc2[r] : acc3[r];
                if (MODE == 2) {
                    float g = 2.f * c;
                    Out[(size_t)m * Ncols + n] = g;
                    if (m != n) { lmax = fmaxf(lmax, g); lmin = fminf(lmin, g); }
                } else {
                    float t2 = fminf(fmaxf(1.f - c * c, 0.f), 1.f);
                    float sine = sqrtf(t2);
                    float phi = (c - TH_C > 0.f) ? (c - MM_C)
                                                 : (c * COS_M - sine * SIN_M);
                    bool onehot = (MODE == 0) ? (lab == n) : (m == n);
                    Out[(size_t)m * Ncols + n] = (onehot ? phi : c) * S_SC;
                }
            }
        }
    }
    if (MODE == 2) {
        __shared__ float smax[256], smin[256];
        smax[threadIdx.x] = lmax;
        smin[threadIdx.x] = lmin;
        __syncthreads();
        for (int s = 128; s > 0; s >>= 1) {
            if (threadIdx.x < s) {
                smax[threadIdx.x] = fmaxf(smax[threadIdx.x], smax[threadIdx.x + s]);
                smin[threadIdx.x] = fminf(smin[threadIdx.x], smin[threadIdx.x + s]);
            }
            __syncthreads();
        }
        if (threadIdx.x == 0) {
            if (smax[0] > -__builtin_inff()) atomicMaxF(&sc[0], smax[0]);
            if (smin[0] < __builtin_inff()) atomicMinF(&sc[1], smin[0]);
        }
    }
}

// --------------------------------------------- offdiag G exp-sum pass
__global__ __launch_bounds__(256) void g_sums(const float* __restrict__ G,
                                              float* __restrict__ sc) {
    __shared__ float r1[256], r2[256];
    const int row = blockIdx.x;
    const int t = threadIdx.x;
    const float vmax = sc[0], gmin = sc[1];
    const float c2 = 2.f - vmax;
    const float coef = -c2 / vmax;
    const float m2 = c2 + (coef >= 0.f ? coef * vmax : coef * gmin);  // max of a2 offdiag
    const float* gr = G + (size_t)row * C_N;
    float s1 = 0.f, s2 = 0.f;
    for (int c = t; c < C_N; c += 256) {
        if (c == row) continue;
        float g = gr[c];
        s1 += expf(g - vmax);                 // S_NEG == 1
        float a2 = c2 * (1.f - g / vmax);
        s2 += expf(a2 - m2);
    }
    r1[t] = s1; r2[t] = s2;
    __syncthreads();
    for (int s = 128; s > 0; s >>= 1) {
        if (t < s) { r1[t] += r1[t + s]; r2[t] += r2[t + s]; }
        __syncthreads();
    }
    if (t == 0) { atomicAdd(&sc[2], r1[0]); atomicAdd(&sc[3], r2[0]); }
}

// ----------------------------------- per-row CE with label smoothing
__global__ __launch_bounds__(256) void ce_rows(const float* __restrict__ logits,
                                               const int* __restrict__ label,
                                               float* __restrict__ acc, int useLabel) {
    __shared__ float srow[C_N];
    __shared__ float red[256];
    __shared__ float sstash;
    const int row = blockIdx.x;
    const int t = threadIdx.x;
    const float* lp = logits + (size_t)row * C_N;
    float lmax = -__builtin_inff();
    for (int i = t; i < C_N; i += 256) {
        float v = lp[i];
        srow[i] = v;
        lmax = fmaxf(lmax, v);
    }
    red[t] = lmax;
    __syncthreads();
    for (int s = 128; s > 0; s >>= 1) {
        if (t < s) red[t] = fmaxf(red[t], red[t + s]);
        __syncthreads();
    }
    const float rmax = red[0];
    __syncthreads();
    float se = 0.f, su = 0.f;
    for (int i = t; i < C_N; i += 256) {
        float v = srow[i];
        se += expf(v - rmax);
        su += v;
    }
    red[t] = se;
    __syncthreads();
    for (int s = 128; s > 0; s >>= 1) {
        if (t < s) red[t] += red[t + s];
        __syncthreads();
    }
    if (t == 0) sstash = red[0];
    __syncthreads();
    red[t] = su;
    __syncthreads();
    for (int s = 128; s > 0; s >>= 1) {
        if (t < s) red[t] += red[t + s];
        __syncthreads();
    }
    if (t == 0) {
        int tgt = useLabel ? label[row] : row;
        float lse = logf(sstash) + rmax;
        float contrib = lse - (1.f - EPS_LS) * srow[tgt]
                            - EPS_LS * (red[0] / (float)C_N);
        atomicAdd(acc, contrib);
    }
}

// ------------------------------------------------- gather center rows
__global__ void gather_rows(const float* __restrict__ CC, const int* __restrict__ label,
                            float* __restrict__ out3) {
    const int i = blockIdx.x;
    const int n = blockIdx.y * 256 + threadIdx.x;
    if (n < C_N) out3[(size_t)i * C_N + n] = CC[(size_t)label[i] * C_N + n];
}

// ------------------------------------------------------------ finalize
__global__ void final_loss(const float* __restrict__ sc, float* __restrict__ out0) {
    if (threadIdx.x == 0 && blockIdx.x == 0) {
        float vmax = sc[0], gmin = sc[1], s1 = sc[2], s2 = sc[3];
        float c2 = 2.f - vmax;
        float coef = -c2 / vmax;
        float m2 = c2 + (coef >= 0.f ? coef * vmax : coef * gmin);
        float lse1 = logf(s1) + vmax;
        float lse2 = logf(s2) + m2;
        float z = lse1 - lse2 + vmax;
        float sp = (z > 30.f) ? z : log1pf(expf(z));
        float loss = sp + 0.5f * (sc[5] / (float)C_N) + 0.5f * (sc[4] / (float)B_N);
        out0[0] = loss;
    }
}

extern "C" void kernel_launch(void* const* d_in, const int* in_sizes, int n_in,
                              void* d_out, int out_size, void* d_ws, size_t ws_size,
                              hipStream_t stream) {
    const float* x = (const float*)d_in[0];
    const float* w = (const float*)d_in[1];
    const int* label = (const int*)d_in[2];
    float* out = (float*)d_out;
    float* ws = (float*)d_ws;

    float* x1 = ws;                              // B*D
    float* wn = x1 + (size_t)B_N * D_N;          // C*D
    float* mx = wn + (size_t)C_N * D_N;          // C*D
    float* cnt = mx + (size_t)C_N * D_N;         // C
    float* CC = cnt + C_N;                       // C*C (reused: G then output_center)
    float* sc = CC + (size_t)C_N * C_N;          // 8 scalars

    float* out_sample = out + 1;
    float* out_center_g = out + 1 + (size_t)B_N * C_N;

    init_ws<<<(C_N * D_N + 255) / 256, 256, 0, stream>>>(mx, cnt, sc);
    normalize_rows<<<B_N, 256, 0, stream>>>(x, x1);
    normalize_rows<<<C_N, 256, 0, stream>>>(w, wn);
    scatter_mean<<<(B_N * D_N + 255) / 256, 256, 0, stream>>>(x1, label, mx, cnt);
    div_mean<<<(C_N * D_N + 255) / 256, 256, 0, stream>>>(mx, cnt);

    dim3 gC((C_N + 255) / 256, (C_N + 31) / 32);
    // G = 2*mean_x @ mean_x^T, fused offdiag max/min
    wmma_gemm<2><<<gC, 256, 0, stream>>>(mx, mx, CC, nullptr, sc, C_N, C_N);
    g_sums<<<C_N, 256, 0, stream>>>(CC, sc);

    // output_center (overwrites G buffer after g_sums consumed it)
    wmma_gemm<1><<<gC, 256, 0, stream>>>(mx, wn, CC, nullptr, sc, C_N, C_N);
    ce_rows<<<C_N, 256, 0, stream>>>(CC, nullptr, &sc[5], 0);
    dim3 gG(B_N, (C_N + 255) / 256);
    gather_rows<<<gG, 256, 0, stream>>>(CC, label, out_center_g);

    // output_sample straight into d_out
    dim3 gS((C_N + 255) / 256, (B_N + 31) / 32);
    wmma_gemm<0><<<gS, 256, 0, stream>>>(x1, wn, out_sample, label, sc, B_N, C_N);
    ce_rows<<<B_N, 256, 0, stream>>>(out_sample, label, &sc[4], 1);

    final_loss<<<1, 1, 0, stream>>>(sc, out);
}